// GAT_85899345920721
// MI455X (gfx1250) — compile-verified
//
#include <hip/hip_runtime.h>
#include <cstddef>

#define N_NODES 10000
#define N_EDGES 100000
#define N_HEADS 8

typedef __attribute__((ext_vector_type(2))) float v2f;
typedef __attribute__((ext_vector_type(8))) float v8f;

// ---------------------------------------------------------------------------
// GEMM: C[M,Nout] = A[M,K] @ B[K,Nout] (+ bias[Nout] if non-null)
// fp32 WMMA (V_WMMA_F32_16X16X4_F32). One wave owns a 16(M) x (16*NT)(N)
// strip; NT is a compile-time template parameter so the k-loop is branch-free
// straight-line code (loads + NT wmma per k-step). 2 waves / block.
// A-frag (16x4 f32): lanes 0-15 hold rows M=0..15 K={k0,k0+1};
//                    lanes 16-31 same rows, K={k0+2,k0+3}.
// B-frag (4x16 f32): vgpr v, lanes 0-15 -> K=k0+v; lanes 16-31 -> K=k0+2+v.
// C/D  (16x16 f32): vgpr j, lanes 0-15 -> M=j; lanes 16-31 -> M=8+j; N=lane&15.
// ---------------------------------------------------------------------------
#define WMMA_F32(Af, Bf, Cf) \
  __builtin_amdgcn_wmma_f32_16x16x4_f32(false, (Af), false, (Bf), (short)0, (Cf), false, false)

template <int NT>
__global__ __launch_bounds__(64) void gemm_wmma_f32(
    const float* __restrict__ A, const float* __restrict__ B,
    const float* __restrict__ bias, float* __restrict__ C,
    int M, int K, int Nout)
{
  const int wave = threadIdx.x >> 5;
  const int lane = threadIdx.x & 31;
  const int m0 = (blockIdx.x * 2 + wave) * 16;
  const int n0 = blockIdx.y * (16 * NT);
  if (m0 >= M) return;                       // wave-uniform exit
  const int half = lane >> 4;
  const int r    = lane & 15;
  const int arow = m0 + r;
  const bool aok = arow < M;
  const int arow_c = aok ? arow : m0;        // clamped (m0 < M always)

  v8f acc0 = {}, acc1 = {}, acc2 = {}, acc3 = {};

  const float* ap = A + (size_t)arow_c * K + 2 * half;
  const float* bp = B + (size_t)(2 * half) * Nout + n0 + r;
  const size_t bstep = (size_t)4 * Nout;

  for (int k0 = 0; k0 < K; k0 += 4) {
    v2f a;
    a[0] = ap[0]; a[1] = ap[1];
    if (!aok) { a[0] = 0.0f; a[1] = 0.0f; }  // predicated select, no branch
    v2f b0, b1, b2, b3;
    b0[0] = bp[0];  b0[1] = bp[Nout];
    if (NT > 1) { b1[0] = bp[16]; b1[1] = bp[16 + Nout]; }
    if (NT > 2) { b2[0] = bp[32]; b2[1] = bp[32 + Nout]; }
    if (NT > 3) { b3[0] = bp[48]; b3[1] = bp[48 + Nout]; }
    acc0 = WMMA_F32(a, b0, acc0);
    if (NT > 1) acc1 = WMMA_F32(a, b1, acc1);
    if (NT > 2) acc2 = WMMA_F32(a, b2, acc2);
    if (NT > 3) acc3 = WMMA_F32(a, b3, acc3);
    ap += 4;
    bp += bstep;
  }

#define STORE_TILE(T, ACC)                                            \
  if (NT > (T)) {                                                     \
    const int n = n0 + 16 * (T) + r;                                  \
    const float bv = bias ? bias[n] : 0.0f;                           \
    _Pragma("unroll")                                                 \
    for (int j = 0; j < 8; ++j) {                                     \
      const int row = m0 + j + 8 * half;                              \
      if (row < M) C[(size_t)row * Nout + n] = (ACC)[j] + bv;         \
    }                                                                 \
  }
  STORE_TILE(0, acc0)
  STORE_TILE(1, acc1)
  STORE_TILE(2, acc2)
  STORE_TILE(3, acc3)
#undef STORE_TILE
}

// ---------------------------------------------------------------------------
// es[n,h] = sum_c H[n,h,c]*a_src[h,c] ; ed likewise
// ---------------------------------------------------------------------------
__global__ void att_coeff(const float* __restrict__ H,
                          const float* __restrict__ a_src,
                          const float* __restrict__ a_dst,
                          float* __restrict__ es, float* __restrict__ ed,
                          int Nn, int heads, int C)
{
  int idx = blockIdx.x * blockDim.x + threadIdx.x;
  if (idx >= Nn * heads) return;
  int n = idx / heads, h = idx - n * heads;
  const float* hp = H + ((size_t)n * heads + h) * C;
  const float* as = a_src + h * C;
  const float* ad = a_dst + h * C;
  float s = 0.f, d = 0.f;
  for (int c = 0; c < C; ++c) {
    float v = hp[c];
    s = fmaf(v, as[c], s);
    d = fmaf(v, ad[c], d);
  }
  es[idx] = s;
  ed[idx] = d;
}

// ---------------------------------------------------------------------------
__global__ void fill_val(float* __restrict__ p, float v, size_t n)
{
  size_t i = (size_t)blockIdx.x * blockDim.x + threadIdx.x;
  if (i < n) p[i] = v;
}

__global__ void fill_bias(float* __restrict__ out, const float* __restrict__ bias,
                          int M, int W)
{
  size_t i = (size_t)blockIdx.x * blockDim.x + threadIdx.x;
  if (i < (size_t)M * W) out[i] = bias[i % W];
}

__global__ void tanh_inplace(float* __restrict__ p, size_t n)
{
  size_t i = (size_t)blockIdx.x * blockDim.x + threadIdx.x;
  if (i < n) p[i] = tanhf(p[i]);
}

// order-preserving float atomic max (works for mixed signs, -inf init)
__device__ inline void atomicMaxFloat(float* addr, float v)
{
  if (v >= 0.0f) atomicMax((int*)addr, __float_as_int(v));
  else           atomicMin((unsigned int*)addr, __float_as_uint(v));
}

// ---------------------------------------------------------------------------
// pass 1: logits e = leaky_relu(es[src]+ed[dst], 0.2); segment max into mx
// edges e >= E_ are the implicit self-loops (src = dst = e - E_)
// ---------------------------------------------------------------------------
__global__ void edge_logits(const int* __restrict__ srcA, const int* __restrict__ dstA,
                            int E_, int Etot,
                            const float* __restrict__ es, const float* __restrict__ ed,
                            float* __restrict__ eb, float* __restrict__ mx, int heads)
{
  int idx = blockIdx.x * blockDim.x + threadIdx.x;
  if (idx >= Etot * heads) return;
  int e = idx / heads, h = idx - e * heads;
  int s = (e < E_) ? srcA[e] : (e - E_);
  int d = (e < E_) ? dstA[e] : (e - E_);
  float v = es[s * heads + h] + ed[d * heads + h];
  v = (v > 0.0f) ? v : 0.2f * v;
  eb[idx] = v;
  atomicMaxFloat(&mx[d * heads + h], v);
}

// pass 2: w = exp(e - mx[dst]); segment sum into dn; store w in-place
__global__ void edge_exp(const int* __restrict__ srcA, const int* __restrict__ dstA,
                         int E_, int Etot,
                         float* __restrict__ eb, const float* __restrict__ mx,
                         float* __restrict__ dn, int heads)
{
  int idx = blockIdx.x * blockDim.x + threadIdx.x;
  if (idx >= Etot * heads) return;
  int e = idx / heads, h = idx - e * heads;
  int d = (e < E_) ? dstA[e] : (e - E_);
  float wv = expf(eb[idx] - mx[d * heads + h]);
  eb[idx] = wv;
  atomicAdd(&dn[d * heads + h], wv);
}

// pass 3: out[dst,h,c] += (w/denom[dst]) * H[src,h,c]; c fastest -> coalesced
__global__ void edge_scatter(const int* __restrict__ srcA, const int* __restrict__ dstA,
                             int E_, int Etot,
                             const float* __restrict__ eb, const float* __restrict__ dn,
                             const float* __restrict__ Hf, float* __restrict__ out,
                             int heads, int C)
{
  size_t idx = (size_t)blockIdx.x * blockDim.x + threadIdx.x;
  size_t total = (size_t)Etot * heads * C;
  if (idx >= total) return;
  int c = (int)(idx % C);
  size_t t = idx / C;
  int h = (int)(t % heads);
  int e = (int)(t / heads);
  int s = (e < E_) ? srcA[e] : (e - E_);
  int d = (e < E_) ? dstA[e] : (e - E_);
  float alpha = eb[(size_t)e * heads + h] / dn[(size_t)d * heads + h];
  atomicAdd(&out[((size_t)d * heads + h) * C + c],
            alpha * Hf[((size_t)s * heads + h) * C + c]);
}

// ---------------------------------------------------------------------------
// GTV prox: z = clip(z + 0.5*(x[i+1]-x[i]), -.5, .5); then x -= D^T z
// ---------------------------------------------------------------------------
__global__ void gtv_z(const float* __restrict__ x, float* __restrict__ z,
                      int Nn, int F)
{
  size_t idx = (size_t)blockIdx.x * blockDim.x + threadIdx.x;
  if (idx >= (size_t)(Nn - 1) * F) return;
  int i = (int)(idx / F), f = (int)(idx % F);
  float dv = x[(size_t)(i + 1) * F + f] - x[(size_t)i * F + f];
  float zv = z[idx] + 0.5f * dv;
  zv = fminf(0.5f, fmaxf(-0.5f, zv));
  z[idx] = zv;
}

__global__ void gtv_x(const float* __restrict__ xin, float* __restrict__ xout,
                      const float* __restrict__ z, int Nn, int F)
{
  size_t idx = (size_t)blockIdx.x * blockDim.x + threadIdx.x;
  if (idx >= (size_t)Nn * F) return;
  int i = (int)(idx / F), f = (int)(idx % F);
  float v = xin[idx];
  if (i < Nn - 1) v += z[(size_t)i * F + f];
  if (i > 0)      v -= z[(size_t)(i - 1) * F + f];
  xout[idx] = v;
}

// ---------------------------------------------------------------------------
// host-side helpers
// ---------------------------------------------------------------------------
static inline unsigned gblk(size_t n) { return (unsigned)((n + 255) / 256); }

static void launch_gemm(hipStream_t s, const float* A, const float* B,
                        const float* bias, float* C, int M, int K, int Nout)
{
  const unsigned gx = (unsigned)((M + 31) / 32);
  if (Nout % 64 == 0) {
    dim3 g(gx, (unsigned)(Nout / 64));
    gemm_wmma_f32<4><<<g, 64, 0, s>>>(A, B, bias, C, M, K, Nout);
  } else if (Nout % 32 == 0) {
    dim3 g(gx, (unsigned)(Nout / 32));
    gemm_wmma_f32<2><<<g, 64, 0, s>>>(A, B, bias, C, M, K, Nout);
  } else {
    dim3 g(gx, (unsigned)((Nout + 15) / 16));
    gemm_wmma_f32<1><<<g, 64, 0, s>>>(A, B, bias, C, M, K, Nout);
  }
}

// full PyG-style GATConv (eval, self-loops implicit)
static void run_gat(hipStream_t s, const int* src, const int* dst,
                    const float* xin, int Cin, int heads, int C,
                    const float* W, const float* a_src, const float* a_dst,
                    const float* bias,
                    float* Hbuf, float* out,
                    float* es, float* ed, float* mx, float* dn, float* eb)
{
  const int Nn = N_NODES;
  const int Wd = heads * C;
  launch_gemm(s, xin, W, nullptr, Hbuf, Nn, Cin, Wd);

  int nh = Nn * heads;
  att_coeff<<<gblk(nh), 256, 0, s>>>(Hbuf, a_src, a_dst, es, ed, Nn, heads, C);
  fill_val<<<gblk(nh), 256, 0, s>>>(mx, -__builtin_inff(), (size_t)nh);
  fill_val<<<gblk(nh), 256, 0, s>>>(dn, 0.0f, (size_t)nh);
  fill_bias<<<gblk((size_t)Nn * Wd), 256, 0, s>>>(out, bias, Nn, Wd);

  const int Etot = N_EDGES + Nn;
  size_t eh = (size_t)Etot * heads;
  edge_logits<<<gblk(eh), 256, 0, s>>>(src, dst, N_EDGES, Etot, es, ed, eb, mx, heads);
  edge_exp<<<gblk(eh), 256, 0, s>>>(src, dst, N_EDGES, Etot, eb, mx, dn, heads);
  size_t ehc = (size_t)Etot * heads * C;
  edge_scatter<<<gblk(ehc), 256, 0, s>>>(src, dst, N_EDGES, Etot, eb, dn, Hbuf, out, heads, C);
}

extern "C" void kernel_launch(void* const* d_in, const int* in_sizes, int n_in,
                              void* d_out, int out_size, void* d_ws, size_t ws_size,
                              hipStream_t stream)
{
  (void)in_sizes; (void)n_in; (void)out_size; (void)ws_size;
  const int Nn = N_NODES;
  const float* d_x = (const float*)d_in[0];
  const int* d_ei  = (const int*)d_in[1];
  const int* src = d_ei;            // edge_index[0]
  const int* dst = d_ei + N_EDGES;  // edge_index[1]
  auto P = [&](int i) { return (const float*)d_in[i]; };

  // ---- workspace carve (floats) ----
  float* w = (float*)d_ws;
  size_t o = 0;
  float* xA = w + o; o += (size_t)Nn * 1024;
  float* xB = w + o; o += (size_t)Nn * 1024;
  float* hB = w + o; o += (size_t)Nn * 1024;
  float* zA = w + o; o += (size_t)(Nn - 1) * 1024;
  float* zB = w + o; o += (size_t)(Nn - 1) * 1024;
  float* eb = w + o; o += (size_t)(N_EDGES + Nn) * N_HEADS;
  float* es = w + o; o += (size_t)Nn * N_HEADS;
  float* ed = w + o; o += (size_t)Nn * N_HEADS;
  float* mx = w + o; o += (size_t)Nn * N_HEADS;
  float* dn = w + o; o += (size_t)Nn * N_HEADS;

  // ---- parameter index map (setup_inputs dict order) ----
  // conv{i} base = 2 + (i-1)*12 ; subs at +4, +8 ; conv8 @ 86 ; lins @ 90..101
  const int convbase[7] = {2, 14, 26, 38, 50, 62, 74};
  const int couts[7]    = {128, 64, 32, 16, 8, 8, 4};
  const int linbase[7]  = {90, 92, 94, 96, 98, -1, 100};
  const int linout[7]   = {1024, 512, 256, 128, 64, 0, 32};
  const int dogtv[7]    = {1, 1, 1, 0, 1, 1, 1};

  // ---- initial GTV (z starts at zero) ----
  {
    size_t zn = (size_t)(Nn - 1) * 512;
    fill_val<<<gblk(zn), 256, 0, stream>>>(zA, 0.0f, zn);
    gtv_z<<<gblk(zn), 256, 0, stream>>>(d_x, zA, Nn, 512);
    size_t xn = (size_t)Nn * 512;
    gtv_x<<<gblk(xn), 256, 0, stream>>>(d_x, xA, zA, Nn, 512);
  }

  float* xc = xA; float* xo = xB;
  float* zc = zA; float* zo = zB;
  int Fcur = 512, zF = 512;

  for (int b = 0; b < 7; ++b) {
    const int C = couts[b];
    const int Wd = N_HEADS * C;
    const int base = convbase[b];

    // x = tanh(gat_conv(x))
    run_gat(stream, src, dst, xc, Fcur, N_HEADS, C,
            P(base), P(base + 1), P(base + 2), P(base + 3),
            hB, xo, es, ed, mx, dn, eb);
    tanh_inplace<<<gblk((size_t)Nn * Wd), 256, 0, stream>>>(xo, (size_t)Nn * Wd);
    { float* t = xc; xc = xo; xo = t; }
    Fcur = Wd;

    // two sub GAT layers, no activation between
    for (int j = 0; j < 2; ++j) {
      const int sb = base + 4 + 4 * j;
      run_gat(stream, src, dst, xc, Fcur, N_HEADS, C,
              P(sb), P(sb + 1), P(sb + 2), P(sb + 3),
              hB, xo, es, ed, mx, dn, eb);
      { float* t = xc; xc = xo; xo = t; }
    }

    // z = z @ W_lin + b_lin
    if (linbase[b] >= 0) {
      launch_gemm(stream, zc, P(linbase[b]), P(linbase[b] + 1), zo,
                  Nn - 1, zF, linout[b]);
      { float* t = zc; zc = zo; zo = t; }
      zF = linout[b];
    }

    // GTV prox on (x, z)
    if (dogtv[b]) {
      size_t zn = (size_t)(Nn - 1) * Fcur;
      gtv_z<<<gblk(zn), 256, 0, stream>>>(xc, zc, Nn, Fcur);
      size_t xn = (size_t)Nn * Fcur;
      gtv_x<<<gblk(xn), 256, 0, stream>>>(xc, xc, zc, Nn, Fcur);
    }

    tanh_inplace<<<gblk((size_t)Nn * Fcur), 256, 0, stream>>>(xc, (size_t)Nn * Fcur);
  }

  // conv8: heads=1, cout=16, concat=False (mean over 1 head == identity)
  run_gat(stream, src, dst, xc, 32, 1, 16,
          P(86), P(87), P(88), P(89),
          hB, (float*)d_out, es, ed, mx, dn, eb);
}